// LatentTCN_29437705847249
// MI455X (gfx1250) — compile-verified
//
#include <hip/hip_runtime.h>
#include <hip/hip_bf16.h>
#include <cstddef>

// ---------------------------------------------------------------------------
// LatentTCN for MI455X (gfx1250, wave32).
//   Stage A: TCN, one WG per batch sample, activations ping-pong in LDS.
//   Stage B: h = relu(latent @ W1 + b1)  (tiny, 32 x 128x128 matvec)
//   Stage C: params = h @ W2 + b2       (530 MB stream -> HBM bound, NT loads)
//   Stage D: out = z @ weight + bias    (per-channel GEMM via V_WMMA_F32_16X16X4_F32)
// ---------------------------------------------------------------------------

#define TCN_L     336
#define TCN_NB    64
#define C_IN      32
#define PRED      96
#define OUT_DIM   (TCN_L * PRED + PRED)   // 32352
#define LAT       128
#define PGD       128
#define BUF_F     (64 * TCN_L)            // 21504 floats per activation buffer
#define WBUF_F    12288                   // largest conv weight (64*64*3)

typedef float v2f __attribute__((ext_vector_type(2)));
typedef float v4f __attribute__((ext_vector_type(4)));
typedef float v8f __attribute__((ext_vector_type(8)));

// ------------------------------- Stage A: TCN ------------------------------

__device__ __forceinline__ void stage_w(const float* __restrict__ g, float* wb, int n, int tid) {
    for (int i = tid; i < n; i += 256) wb[i] = g[i];
}

__device__ __forceinline__ void conv_relu(const float* __restrict__ in, float* __restrict__ out,
                                          const float* __restrict__ wb, const float* __restrict__ bias,
                                          int c_out, int c_in, int dil, int tid) {
    const int LT = TCN_L / 4;  // 84 tiles of 4 outputs
    for (int t = tid; t < c_out * LT; t += 256) {
        int co = t / LT;
        int l0 = (t - co * LT) * 4;
        float bv = bias[co];
        float acc[4] = {bv, bv, bv, bv};
        const float* wr = wb + co * c_in * 3;
        for (int ci = 0; ci < c_in; ci++) {
            float w0 = wr[3 * ci + 0], w1 = wr[3 * ci + 1], w2 = wr[3 * ci + 2];
            const float* row = in + ci * TCN_L;
#pragma unroll
            for (int j = 0; j < 4; j++) {
                int l = l0 + j;
                int i0 = l - 2 * dil, i1 = l - dil;
                float a0 = (i0 >= 0) ? row[i0] : 0.f;
                float a1 = (i1 >= 0) ? row[i1] : 0.f;
                acc[j] += w0 * a0 + w1 * a1 + w2 * row[l];
            }
        }
        float* orow = out + co * TCN_L + l0;
#pragma unroll
        for (int j = 0; j < 4; j++) orow[j] = fmaxf(acc[j], 0.f);
    }
}

__device__ __forceinline__ void run_block(const float* cur, float* tmp, float* out,
                                          const float* w1, const float* b1,
                                          const float* w2, const float* b2,
                                          const float* wd, const float* bd,
                                          int c_out, int c_in, int dil, float* wb, int tid) {
    stage_w(w1, wb, c_out * c_in * 3, tid);
    __syncthreads();
    conv_relu(cur, tmp, wb, b1, c_out, c_in, dil, tid);
    __syncthreads();
    stage_w(w2, wb, c_out * c_out * 3, tid);
    __syncthreads();
    conv_relu(tmp, out, wb, b2, c_out, c_out, dil, tid);
    __syncthreads();
    if (wd != nullptr) {
        stage_w(wd, wb, c_out * c_in, tid);
        __syncthreads();
        for (int t = tid; t < c_out * TCN_L; t += 256) {
            int co = t / TCN_L, l = t - co * TCN_L;
            float r = bd[co];
            const float* wr = wb + co * c_in;
            for (int ci = 0; ci < c_in; ci++) r += wr[ci] * cur[ci * TCN_L + l];
            out[t] = fmaxf(out[t] + r, 0.f);
        }
    } else {
        for (int t = tid; t < c_out * TCN_L; t += 256)
            out[t] = fmaxf(out[t] + cur[t], 0.f);
    }
    __syncthreads();
}

__global__ __launch_bounds__(256) void tcn_kernel(
    const float* __restrict__ X,
    const float* w1_0, const float* b1_0, const float* w2_0, const float* b2_0,
    const float* wd_0, const float* bd_0,
    const float* w1_1, const float* b1_1, const float* w2_1, const float* b2_1,
    const float* w1_2, const float* b1_2, const float* w2_2, const float* b2_2,
    const float* wd_2, const float* bd_2,
    float* __restrict__ zout) {
    extern __shared__ float smem[];
    float* A  = smem;
    float* B  = smem + BUF_F;
    float* Cb = smem + 2 * BUF_F;
    float* wb = smem + 3 * BUF_F;
    const int tid = threadIdx.x;
    const int n = blockIdx.x;

    // transpose X[n] (L, C_IN) -> LDS [c][l]
    for (int i = tid; i < C_IN * TCN_L; i += 256) {
        int c = i / TCN_L, l = i - c * TCN_L;
        A[i] = X[((size_t)n * TCN_L + l) * C_IN + c];
    }
    __syncthreads();

    run_block(A,  B, Cb, w1_0, b1_0, w2_0, b2_0, wd_0, bd_0, 64, 32, 1, wb, tid);
    run_block(Cb, B, A,  w1_1, b1_1, w2_1, b2_1, nullptr, nullptr, 64, 64, 2, wb, tid);
    run_block(A,  B, Cb, w1_2, b1_2, w2_2, b2_2, wd_2, bd_2, 32, 64, 4, wb, tid);

    // z[n][c][l]  (layout matches i = c*L + l)
    for (int i = tid; i < C_IN * TCN_L; i += 256)
        zout[(size_t)n * C_IN * TCN_L + i] = Cb[i];
}

// --------------------------- Stage B: h = relu(latent@W1+b1) ----------------

__global__ __launch_bounds__(128) void hyper_h_kernel(
    const float* __restrict__ latent, const float* __restrict__ W1,
    const float* __restrict__ b1h, float* __restrict__ hbuf) {
    __shared__ float ls[LAT];
    int c = blockIdx.x;
    int t = threadIdx.x;
    ls[t] = latent[c * LAT + t];
    __syncthreads();
    float acc = b1h[c * PGD + t];
    const float* Wc = W1 + (size_t)c * LAT * PGD;
    for (int d = 0; d < LAT; d++) acc += ls[d] * Wc[d * PGD + t];
    hbuf[c * PGD + t] = fmaxf(acc, 0.f);
}

// --------------- Stage C: params = h @ W2 + b2 (HBM-bound stream) -----------

__global__ __launch_bounds__(256) void hyper_params_kernel(
    const float* __restrict__ W2, const float* __restrict__ b2h,
    const float* __restrict__ hbuf, float* __restrict__ params) {
    __shared__ float hs[PGD];
    const int c = blockIdx.y;
    if (threadIdx.x < PGD) hs[threadIdx.x] = hbuf[c * PGD + threadIdx.x];
    __syncthreads();
    int o4 = (blockIdx.x * 256 + threadIdx.x) * 4;
    if (o4 >= OUT_DIM) return;
    const v4f* Wp = reinterpret_cast<const v4f*>(W2 + (size_t)c * PGD * OUT_DIM);
    v4f acc = *reinterpret_cast<const v4f*>(b2h + (size_t)c * OUT_DIM + o4);
    const size_t col = (size_t)(o4 >> 2);
    const size_t rstride = OUT_DIM / 4;  // 8088 float4 per row
#pragma unroll 4
    for (int j = 0; j < PGD; j++) {
        v4f w = __builtin_nontemporal_load(&Wp[(size_t)j * rstride + col]);  // one-shot 530MB stream: NT
        float hv = hs[j];
        acc.x += hv * w.x;
        acc.y += hv * w.y;
        acc.z += hv * w.z;
        acc.w += hv * w.w;
    }
    *reinterpret_cast<v4f*>(params + (size_t)c * OUT_DIM + o4) = acc;
}

// ------- Stage D: per-channel GEMM via V_WMMA_F32_16X16X4_F32 (fp32) --------
// A = z_c (64x336), B = weight_c (336x96), C = 16x16 tiles; grid (4*6, 32ch).

__global__ __launch_bounds__(32) void proj_wmma_kernel(
    const float* __restrict__ z, const float* __restrict__ params,
    float* __restrict__ out) {
    const int c    = blockIdx.y;
    const int tile = blockIdx.x;         // 0..23
    const int mt = tile / 6, nt = tile - mt * 6;
    const int m0 = mt * 16, p0 = nt * 16;
    const int lane = threadIdx.x;
    const int half = lane >> 4;          // 0: K,K+1  1: K+2,K+3 (ISA 16x4 A layout)
    const int r    = lane & 15;

    const float* Wc = params + (size_t)c * OUT_DIM;      // weight[c]: 336 x 96 row-major
    const float* zr = z + (size_t)(m0 + r) * (C_IN * TCN_L) + (size_t)c * TCN_L;

    v8f acc = {};
    for (int k = 0; k < TCN_L; k += 4) {
        const int kk = k + 2 * half;
        v2f a, b;
        a.x = zr[kk];
        a.y = zr[kk + 1];
        b.x = Wc[kk * PRED + p0 + r];
        b.y = Wc[(kk + 1) * PRED + p0 + r];
        acc = __builtin_amdgcn_wmma_f32_16x16x4_f32(
            /*neg_a=*/false, a, /*neg_b=*/false, b,
            /*c_mod=*/(short)0, acc, /*reuse_a=*/false, /*reuse_b=*/false);
    }

    const float bias = Wc[TCN_L * PRED + p0 + r];
    const int p = p0 + r;
#pragma unroll
    for (int v = 0; v < 8; v++) {
        int n = m0 + v + half * 8;                       // ISA 16x16 C/D layout
        out[(size_t)n * (PRED * C_IN) + p * C_IN + c] = acc[v] + bias;
    }
}

// ------------------------------- host launch --------------------------------

extern "C" void kernel_launch(void* const* d_in, const int* in_sizes, int n_in,
                              void* d_out, int out_size, void* d_ws, size_t ws_size,
                              hipStream_t stream) {
    const float* X = (const float*)d_in[0];
    const float *w1[3], *b1[3], *w2[3], *b2[3], *wd[3], *bd[3];

    // tcn_params flattening: insertion order (w1,b1,w2,b2[,wd,bd]) vs
    // sorted-key pytree order (b1,b2[,bd],w1,w2[,wd]); detect via in_sizes[1].
    if (in_sizes[1] == 6144) {  // w1_0 first -> insertion order
        w1[0] = (const float*)d_in[1];  b1[0] = (const float*)d_in[2];
        w2[0] = (const float*)d_in[3];  b2[0] = (const float*)d_in[4];
        wd[0] = (const float*)d_in[5];  bd[0] = (const float*)d_in[6];
        w1[1] = (const float*)d_in[7];  b1[1] = (const float*)d_in[8];
        w2[1] = (const float*)d_in[9];  b2[1] = (const float*)d_in[10];
        wd[1] = nullptr;                bd[1] = nullptr;
        w1[2] = (const float*)d_in[11]; b1[2] = (const float*)d_in[12];
        w2[2] = (const float*)d_in[13]; b2[2] = (const float*)d_in[14];
        wd[2] = (const float*)d_in[15]; bd[2] = (const float*)d_in[16];
    } else {                    // sorted dict keys
        b1[0] = (const float*)d_in[1];  b2[0] = (const float*)d_in[2];
        bd[0] = (const float*)d_in[3];  w1[0] = (const float*)d_in[4];
        w2[0] = (const float*)d_in[5];  wd[0] = (const float*)d_in[6];
        b1[1] = (const float*)d_in[7];  b2[1] = (const float*)d_in[8];
        w1[1] = (const float*)d_in[9];  w2[1] = (const float*)d_in[10];
        wd[1] = nullptr;                bd[1] = nullptr;
        b1[2] = (const float*)d_in[11]; b2[2] = (const float*)d_in[12];
        bd[2] = (const float*)d_in[13]; w1[2] = (const float*)d_in[14];
        w2[2] = (const float*)d_in[15]; wd[2] = (const float*)d_in[16];
    }
    const float* latent = (const float*)d_in[17];
    const float* W1h    = (const float*)d_in[18];
    const float* b1h    = (const float*)d_in[19];
    const float* W2h    = (const float*)d_in[20];
    const float* b2h    = (const float*)d_in[21];

    float* out  = (float*)d_out;
    float* zout = out + (size_t)TCN_NB * PRED * C_IN;        // z region of d_out
    float* hbuf = (float*)d_ws;                              // 32*128 floats
    float* params = hbuf + C_IN * PGD;                       // 32*32352 floats

    const size_t tcnLds = (size_t)(3 * BUF_F + WBUF_F) * sizeof(float);  // 300 KB < 320 KB/WGP
    tcn_kernel<<<dim3(TCN_NB), dim3(256), tcnLds, stream>>>(
        X, w1[0], b1[0], w2[0], b2[0], wd[0], bd[0],
           w1[1], b1[1], w2[1], b2[1],
           w1[2], b1[2], w2[2], b2[2], wd[2], bd[2], zout);

    hyper_h_kernel<<<dim3(C_IN), dim3(128), 0, stream>>>(latent, W1h, b1h, hbuf);

    const int nblk = (OUT_DIM / 4 + 255) / 256;              // 32 blocks/channel
    hyper_params_kernel<<<dim3(nblk, C_IN), dim3(256), 0, stream>>>(W2h, b2h, hbuf, params);

    proj_wmma_kernel<<<dim3(24, C_IN), dim3(32), 0, stream>>>(zout, params, out);
}